// Encoder_8435315769854
// MI455X (gfx1250) — compile-verified
//
#include <hip/hip_runtime.h>
#include <math.h>

// ---- problem constants (match reference) ----
constexpr int  cB = 16, cN = 256, cP = 48, cD = 128, cE = 12, cF = 64, cD2 = 64;
constexpr long cNPD  = (long)cN * cP * cD;        // 1,572,864
constexpr long cBNPD = (long)cB * cN * cP * cD;   // 25,165,824

typedef float v2f __attribute__((ext_vector_type(2)));
typedef float v8f __attribute__((ext_vector_type(8)));

// ---- GEMM epilogue flags ----
#define G_RELUA 1   // relu on A-operand
#define G_RELUO 2   // relu on output (after bias)
#define G_ACC   4   // C += result
#define G_RES   8   // += residual (after bias/relu)
#define G_BIAS 16   // += bias[col]

struct GemmArgs {
  const float* A; const float* Bm; float* C; const float* R; const float* bias;
  int M, Ncol, K;
  int lda; int a_div; long a_hi; long a_lo;    // A batch off = (bs/a_div)*a_hi + (bs%a_div)*a_lo
  int b_rs; int b_cs; int b_div; long b_hi; long b_lo; // B[k,n] at k*b_rs + n*b_cs (+batch)
  int ldc; long c_hi;                           // C contiguous per bs
  int ldr; int r_div; long r_hi; long r_lo;     // residual strides
};

// One wave computes a 32x64 output strip: 2 row-tiles x 4 col-tiles, 8 accumulators.
// K-loop is software-pipelined: iteration k+4's 6 fragments are issued before the
// 8 WMMAs of iteration k, so a full WMMA step hides the load latency.
// f32 WMMA fragment layout (ISA 7.12.2): A 16x4: lane&15=row, lane>>4 selects K-pair.
// C/D 16x16: lane&15=col, vgpr r = row (lane>>4 adds 8).  B 4x16 mirrors A.
// BT=true : b_rs==1 (transposed weights) -> B pair is one aligned v2f load.
// BT=false: b_cs==1 (row-major activations) -> two b32 loads g.b_rs apart.
template<bool BT, int FLAGS>
__global__ __launch_bounds__(128)
void gemm_wmma_f32(GemmArgs g) {
  const int lane = threadIdx.x & 31;
  const int wave = threadIdx.x >> 5;
  const int m0 = blockIdx.y * 128 + wave * 32;
  if (m0 >= g.M) return;                       // wave-uniform exit (EXEC stays all-1)
  const bool has2 = (m0 + 16) < g.M;
  const int n0 = blockIdx.x * 64;
  const int bs = blockIdx.z;
  const int lh = lane & 15;
  const int kh = lane >> 4;

  const long aoff = (long)(bs / g.a_div) * g.a_hi + (long)(bs % g.a_div) * g.a_lo;
  const float* A0 = g.A + aoff + (long)(m0 + lh) * g.lda + 2 * kh;
  const int  r2   = has2 ? (m0 + 16 + lh) : (g.M - 1);   // clamp keeps loads in-bounds
  const float* A1 = g.A + aoff + (long)r2 * g.lda + 2 * kh;
  const float* Bb = g.Bm + (long)(bs / g.b_div) * g.b_hi + (long)(bs % g.b_div) * g.b_lo
                         + 2 * kh * (BT ? 1 : g.b_rs);
  float* Cb = g.C + (long)bs * g.c_hi;

  v8f a00 = {}, a01 = {}, a02 = {}, a03 = {};
  v8f a10 = {}, a11 = {}, a12 = {}, a13 = {};

#define LOAD_FRAGS(KK, AV0, AV1, B0, B1, B2, B3) { \
    AV0 = *(const v2f*)(A0 + (KK)); \
    AV1 = *(const v2f*)(A1 + (KK)); \
    const float* bp = Bb + (BT ? (long)(KK) : (long)(KK) * g.b_rs); \
    if (BT) { \
      B0 = *(const v2f*)(bp + (long)(n0 +  0 + lh) * g.b_cs); \
      B1 = *(const v2f*)(bp + (long)(n0 + 16 + lh) * g.b_cs); \
      B2 = *(const v2f*)(bp + (long)(n0 + 32 + lh) * g.b_cs); \
      B3 = *(const v2f*)(bp + (long)(n0 + 48 + lh) * g.b_cs); \
    } else { \
      const float* q0 = bp + (n0 +  0 + lh); B0.x = q0[0]; B0.y = q0[g.b_rs]; \
      const float* q1 = bp + (n0 + 16 + lh); B1.x = q1[0]; B1.y = q1[g.b_rs]; \
      const float* q2 = bp + (n0 + 32 + lh); B2.x = q2[0]; B2.y = q2[g.b_rs]; \
      const float* q3 = bp + (n0 + 48 + lh); B3.x = q3[0]; B3.y = q3[g.b_rs]; \
    } \
  }

#define WMMA8(AV0, AV1, B0, B1, B2, B3) { \
    v2f x0 = AV0, x1 = AV1; \
    if (FLAGS & G_RELUA) { \
      x0.x = fmaxf(x0.x, 0.f); x0.y = fmaxf(x0.y, 0.f); \
      x1.x = fmaxf(x1.x, 0.f); x1.y = fmaxf(x1.y, 0.f); \
    } \
    a00 = __builtin_amdgcn_wmma_f32_16x16x4_f32(false, x0, false, B0, (short)0, a00, false, false); \
    a10 = __builtin_amdgcn_wmma_f32_16x16x4_f32(false, x1, false, B0, (short)0, a10, false, false); \
    a01 = __builtin_amdgcn_wmma_f32_16x16x4_f32(false, x0, false, B1, (short)0, a01, false, false); \
    a11 = __builtin_amdgcn_wmma_f32_16x16x4_f32(false, x1, false, B1, (short)0, a11, false, false); \
    a02 = __builtin_amdgcn_wmma_f32_16x16x4_f32(false, x0, false, B2, (short)0, a02, false, false); \
    a12 = __builtin_amdgcn_wmma_f32_16x16x4_f32(false, x1, false, B2, (short)0, a12, false, false); \
    a03 = __builtin_amdgcn_wmma_f32_16x16x4_f32(false, x0, false, B3, (short)0, a03, false, false); \
    a13 = __builtin_amdgcn_wmma_f32_16x16x4_f32(false, x1, false, B3, (short)0, a13, false, false); \
  }

  // software pipeline: fragments for step k are in flight while step k-4 computes
  v2f ca0, ca1, cb0, cb1, cb2, cb3;
  LOAD_FRAGS(0, ca0, ca1, cb0, cb1, cb2, cb3)
  for (int k = 4; k < g.K; k += 4) {
    v2f na0, na1, nb0, nb1, nb2, nb3;
    LOAD_FRAGS(k, na0, na1, nb0, nb1, nb2, nb3)
    WMMA8(ca0, ca1, cb0, cb1, cb2, cb3)
    ca0 = na0; ca1 = na1; cb0 = nb0; cb1 = nb1; cb2 = nb2; cb3 = nb3;
  }
  WMMA8(ca0, ca1, cb0, cb1, cb2, cb3)
#undef LOAD_FRAGS
#undef WMMA8

  const float* Rb = nullptr;
  if (FLAGS & G_RES)
    Rb = g.R + (long)(bs / g.r_div) * g.r_hi + (long)(bs % g.r_div) * g.r_lo;

#define EPILOG(J, ACC, MB) { \
    const int col = n0 + (J)*16 + lh; \
    const float bia = (FLAGS & G_BIAS) ? g.bias[col] : 0.f; \
    _Pragma("unroll") \
    for (int r = 0; r < 8; ++r) { \
      const int row = (MB) + r + 8 * kh; \
      const long ci = (long)row * g.ldc + col; \
      float v = ACC[r]; \
      if (FLAGS & G_ACC) v += Cb[ci]; \
      v += bia; \
      if (FLAGS & G_RELUO) v = fmaxf(v, 0.f); \
      if (FLAGS & G_RES) v += Rb[(long)row * g.ldr + col]; \
      Cb[ci] = v; \
    } \
  }
  EPILOG(0, a00, m0)
  EPILOG(1, a01, m0)
  EPILOG(2, a02, m0)
  EPILOG(3, a03, m0)
  if (has2) {
    EPILOG(0, a10, m0 + 16)
    EPILOG(1, a11, m0 + 16)
    EPILOG(2, a12, m0 + 16)
    EPILOG(3, a13, m0 + 16)
  }
#undef EPILOG
}

// pooled[b,m,:] = mean over s of X[b,m,s,:]  (strided logical layout)
__global__ void pool_kernel(const float* X, float* pooled, int S, long m_stride, long s_stride, int M) {
  const int bm = blockIdx.x, b = bm / M, m = bm % M, d = threadIdx.x;
  const float* p = X + (long)b * cNPD + (long)m * m_stride + d;
  float s = 0.f;
  for (int i = 0; i < S; ++i) s += p[(long)i * s_stride];
  pooled[(long)bm * cD + d] = s / (float)S;
}

// e1/e2 = normalize(relu(pooled @ w.T)) ; one wave per (b,m)
__global__ void embed_kernel(const float* pooled, const float* w1, const float* w2,
                             float* e1, float* e2) {
  const int bm = blockIdx.x, t = threadIdx.x;
  __shared__ float v[24];
  __shared__ float nrm[2];
  const float* pr = pooled + (long)bm * cD;
  if (t < 24) {
    const float* w = (t < cE) ? (w1 + (long)t * cD) : (w2 + (long)(t - cE) * cD);
    float s = 0.f;
    for (int d = 0; d < cD; ++d) s += pr[d] * w[d];
    v[t] = fmaxf(s, 0.f);
  }
  __syncthreads();
  if (t < 2) {
    float s = 0.f;
    for (int j = 0; j < cE; ++j) { float x = v[t * cE + j]; s += x * x; }
    nrm[t] = sqrtf(s) + 0.001f;
  }
  __syncthreads();
  if (t < cE)            e1[(long)bm * cE + t]        = v[t] / nrm[0];
  else if (t < 2 * cE)   e2[(long)bm * cE + (t - cE)] = v[t] / nrm[1];
}

// A[b,i,j] = relu(e1[b,i,:] . e2[b,j,:])
__global__ void adj_kernel(const float* e1, const float* e2, float* A, int M, long tot) {
  const long idx = (long)blockIdx.x * 256 + threadIdx.x;
  if (idx >= tot) return;
  const int j = (int)(idx % M); const long q = idx / M;
  const int i = (int)(q % M);   const int b = (int)(q / M);
  const float* p1 = e1 + ((long)b * M + i) * cE;
  const float* p2 = e2 + ((long)b * M + j) * cE;
  float s = 0.f;
  for (int e = 0; e < cE; ++e) s += p1[e] * p2[e];
  A[idx] = fmaxf(s, 0.f);
}

__global__ void adj_sums(const float* A, float* rs, float* cs, int M) {
  const int idx = blockIdx.x * 128 + threadIdx.x;
  const int tot = cB * M;
  if (idx >= 2 * tot) return;
  const int which = idx / tot, bm = idx % tot, b = bm / M, i = bm % M;
  const float* Ab = A + (long)b * M * M;
  float s = 0.f;
  if (which == 0) { for (int j = 0; j < M; ++j) s += Ab[(long)i * M + j]; rs[bm] = s; }
  else            { for (int j = 0; j < M; ++j) s += Ab[(long)j * M + i]; cs[bm] = s; }
}

// A1[b,i,j] = A[b,i,j]/(rowsum[b,i]+1e-3) ; A2[b,i,j] = A[b,j,i]/(colsum[b,i]+1e-3)
__global__ void adj_norm(const float* A, const float* rs, const float* cs,
                         float* A1, float* A2, int M, long tot) {
  const long idx = (long)blockIdx.x * 256 + threadIdx.x;
  if (idx >= tot) return;
  const int j = (int)(idx % M); const long q = idx / M;
  const int i = (int)(q % M);   const int b = (int)(q / M);
  A1[idx] = A[idx] / (rs[(long)b * M + i] + 0.001f);
  A2[idx] = A[((long)b * M + j) * M + i] / (cs[(long)b * M + i] + 0.001f);
}

// joint LayerNorm over [M,D] per (b,s); in-place
__global__ __launch_bounds__(256)
void ln_kernel(float* y, const float* g, const float* bb, int MD) {
  const long base = (long)blockIdx.x * MD;
  const int t = threadIdx.x;
  float s = 0.f, ss = 0.f;
  for (int i = t; i < MD; i += 256) { float v = y[base + i]; s += v; ss += v * v; }
  for (int o = 16; o > 0; o >>= 1) { s += __shfl_down(s, o, 32); ss += __shfl_down(ss, o, 32); }
  __shared__ float rs[8], rq[8];
  __shared__ float mu, inv;
  const int w = t >> 5;
  if ((t & 31) == 0) { rs[w] = s; rq[w] = ss; }
  __syncthreads();
  if (t == 0) {
    float S1 = 0.f, S2 = 0.f;
    for (int i = 0; i < 8; ++i) { S1 += rs[i]; S2 += rq[i]; }
    const float m = S1 / MD;
    mu = m; inv = rsqrtf(S2 / MD - m * m + 1e-5f);
  }
  __syncthreads();
  for (int i = t; i < MD; i += 256)
    y[base + i] = (y[base + i] - mu) * inv * g[i] + bb[i];
}

// out_row = sigmoid(x_row . w + b) * x_row   (with input/output transpose strides)
__global__ __launch_bounds__(128)
void gate_kernel(const float* in, float* out, const float* gw, const float* gb,
                 int n1, int n2, long is1, long is2, long os1, long os2) {
  const long r = blockIdx.x;
  const int t = threadIdx.x;
  const int j = (int)(r % n2); const long q = r / n2;
  const int i = (int)(q % n1); const int b = (int)(q / n1);
  const float* ip = in + (long)b * cNPD + (long)i * is1 + (long)j * is2;
  const float v = ip[t];
  float pp = v * gw[t];
  for (int o = 16; o > 0; o >>= 1) pp += __shfl_down(pp, o, 32);
  __shared__ float ws4[4];
  __shared__ float gs;
  if ((t & 31) == 0) ws4[t >> 5] = pp;
  __syncthreads();
  if (t == 0) {
    const float s = ws4[0] + ws4[1] + ws4[2] + ws4[3] + gb[0];
    gs = 1.f / (1.f + expf(-s));
  }
  __syncthreads();
  out[(long)b * cNPD + (long)i * os1 + (long)j * os2 + t] = gs * v;
}

// ==================== host side ====================

struct GragP {
  const float *e1_w, *e2_w, *W, *ffn_w1, *ffn_b1, *ffn_w2, *ffn_b2,
              *ffn2_w1, *ffn2_b1, *ffn2_w2, *ffn2_b2, *ln_g, *ln_b;
};
struct Ws {
  float *act, *y, *u, *U, *h, *pooled, *e1, *e2, *A1, *A2, *rsum, *csum;
};

static inline const float* FP(void* const* d, int i) { return (const float*)d[i]; }

static GragP grag_ins(void* const* d, int& i) {   // insertion-order flattening
  GragP g;
  g.e1_w = FP(d,i++); g.e2_w = FP(d,i++); g.W = FP(d,i++);
  g.ffn_w1 = FP(d,i++); g.ffn_b1 = FP(d,i++); g.ffn_w2 = FP(d,i++); g.ffn_b2 = FP(d,i++);
  g.ffn2_w1 = FP(d,i++); g.ffn2_b1 = FP(d,i++); g.ffn2_w2 = FP(d,i++); g.ffn2_b2 = FP(d,i++);
  g.ln_g = FP(d,i++); g.ln_b = FP(d,i++);
  return g;
}
static GragP grag_srt(void* const* d, int& i) {   // jax sorted-keys flattening
  GragP g;
  g.W = FP(d,i++); g.e1_w = FP(d,i++); g.e2_w = FP(d,i++);
  g.ffn2_b1 = FP(d,i++); g.ffn2_b2 = FP(d,i++); g.ffn2_w1 = FP(d,i++); g.ffn2_w2 = FP(d,i++);
  g.ffn_b1 = FP(d,i++); g.ffn_b2 = FP(d,i++); g.ffn_w1 = FP(d,i++); g.ffn_w2 = FP(d,i++);
  g.ln_b = FP(d,i++); g.ln_g = FP(d,i++);
  return g;
}

static inline void gemm(hipStream_t st,
  const float* A, int lda, int a_div, long a_hi, long a_lo,
  const float* Bm, int b_rs, int b_cs, int b_div, long b_hi, long b_lo,
  float* C, int ldc, long c_hi,
  const float* R, int ldr, int r_div, long r_hi, long r_lo,
  const float* bias, int M, int Ncol, int K, int BS, int flags)
{
  GemmArgs g;
  g.A = A; g.Bm = Bm; g.C = C; g.R = R; g.bias = bias;
  g.M = M; g.Ncol = Ncol; g.K = K;
  g.lda = lda; g.a_div = a_div; g.a_hi = a_hi; g.a_lo = a_lo;
  g.b_rs = b_rs; g.b_cs = b_cs; g.b_div = b_div; g.b_hi = b_hi; g.b_lo = b_lo;
  g.ldc = ldc; g.c_hi = c_hi;
  g.ldr = ldr; g.r_div = (r_div > 0 ? r_div : 1); g.r_hi = r_hi; g.r_lo = r_lo;
  dim3 grid(Ncol / 64, (M + 127) / 128, BS);
  const dim3 blk(128);
  const bool bt = (b_rs == 1);
  if (bt) {
    if (flags == (G_RELUA | G_BIAS | G_RELUO))
      gemm_wmma_f32<true,  G_RELUA | G_BIAS | G_RELUO><<<grid, blk, 0, st>>>(g);
    else
      gemm_wmma_f32<true,  G_BIAS | G_RES><<<grid, blk, 0, st>>>(g);
  } else {
    if (flags == 0)          gemm_wmma_f32<false, 0><<<grid, blk, 0, st>>>(g);
    else if (flags == G_RES) gemm_wmma_f32<false, G_RES><<<grid, blk, 0, st>>>(g);
    else                     gemm_wmma_f32<false, G_ACC><<<grid, blk, 0, st>>>(g);
  }
}

// One Pruned_GrAG: Xin logical [B,M,S,D] with given strides; result -> yOut [bs,M,D]
static void run_grag(hipStream_t st, const float* Xin, int M, int S,
                     int m_stride, long s_stride, const GragP& p,
                     float* yOut, float* Araw, const Ws& w)
{
  const int BS = cB * S;
  pool_kernel<<<cB * M, cD, 0, st>>>(Xin, w.pooled, S, m_stride, s_stride, M);
  embed_kernel<<<cB * M, 32, 0, st>>>(w.pooled, p.e1_w, p.e2_w, w.e1, w.e2);
  const long totA = (long)cB * M * M;
  adj_kernel<<<(unsigned)((totA + 255) / 256), 256, 0, st>>>(w.e1, w.e2, Araw, M, totA);
  adj_sums<<<(cB * M * 2 + 127) / 128, 128, 0, st>>>(Araw, w.rsum, w.csum, M);
  adj_norm<<<(unsigned)((totA + 255) / 256), 256, 0, st>>>(Araw, w.rsum, w.csum, w.A1, w.A2, M, totA);

  // U[:, :, k*F:(k+1)*F] = X @ W[k]   (K = D)
  for (int kk = 0; kk < 3; ++kk)
    gemm(st, Xin, m_stride, S, cNPD, s_stride,
         p.W + (long)kk * cD * cF, cF, 1, 1, 0, 0,
         w.U + kk * cF, 3 * cF, (long)M * 3 * cF,
         nullptr, 0, 1, 0, 0, nullptr, M, cF, cD, BS, 0);
  // h = A1 @ U1 + U0    (K = M)
  gemm(st, w.A1, M, S, (long)M * M, 0,
       w.U + cF, 3 * cF, 1, 1, (long)M * 3 * cF, 0,
       w.h, cF, (long)M * cF,
       w.U, 3 * cF, 1, (long)M * 3 * cF, 0, nullptr, M, cF, M, BS, G_RES);
  // h += A2 @ U2
  gemm(st, w.A2, M, S, (long)M * M, 0,
       w.U + 2 * cF, 3 * cF, 1, 1, (long)M * 3 * cF, 0,
       w.h, cF, (long)M * cF,
       nullptr, 0, 1, 0, 0, nullptr, M, cF, M, BS, G_ACC);
  // u = relu(relu(h) @ ffn_w1.T + b1)
  gemm(st, w.h, cF, 1, (long)M * cF, 0,
       p.ffn_w1, 1, cF, 1, 0, 0,
       w.u, cD, (long)M * cD,
       nullptr, 0, 1, 0, 0, p.ffn_b1, M, cD, cF, BS, G_RELUA | G_BIAS | G_RELUO);
  // y = u @ ffn_w2.T + b2 + X
  gemm(st, w.u, cD, 1, (long)M * cD, 0,
       p.ffn_w2, 1, cD, 1, 0, 0,
       yOut, cD, (long)M * cD,
       Xin, m_stride, S, cNPD, s_stride, p.ffn_b2, M, cD, cD, BS, G_BIAS | G_RES);
  ln_kernel<<<BS, 256, 0, st>>>(yOut, p.ln_g, p.ln_b, M * cD);
  // t = relu(relu(y) @ ffn2_w1.T + b1)   (reuse h buffer, N = D2)
  gemm(st, yOut, cD, 1, (long)M * cD, 0,
       p.ffn2_w1, 1, cD, 1, 0, 0,
       w.h, cD2, (long)M * cD2,
       nullptr, 0, 1, 0, 0, p.ffn2_b1, M, cD2, cD, BS, G_RELUA | G_BIAS | G_RELUO);
  // y = y + t @ ffn2_w2.T + b2
  gemm(st, w.h, cD2, 1, (long)M * cD2, 0,
       p.ffn2_w2, 1, cD2, 1, 0, 0,
       yOut, cD, (long)M * cD,
       yOut, cD, 1, (long)M * cD, 0, p.ffn2_b2, M, cD, cD2, BS, G_BIAS | G_RES);
  ln_kernel<<<BS, 256, 0, st>>>(yOut, p.ln_g, p.ln_b, M * cD);
}

extern "C" void kernel_launch(void* const* d_in, const int* in_sizes, int n_in,
                              void* d_out, int out_size, void* d_ws, size_t ws_size,
                              hipStream_t stream) {
  (void)n_in; (void)out_size; (void)ws_size;
  // Detect flattening scheme: insertion-order puts x (B*N*P*D elements) first.
  const bool ins = (in_sizes[0] == (int)cBNPD);
  const float* x = nullptr;
  GragP g[2][2];                    // [block][0=intra,1=inter]
  const float *bgw[2], *bgb[2], *fgw, *fgb;
  int i = 0;
  if (ins) {
    x = FP(d_in, 0); i = 1;
    for (int b = 0; b < 2; ++b) {
      g[b][0] = grag_ins(d_in, i);
      g[b][1] = grag_ins(d_in, i);
      bgw[b] = FP(d_in, i++); bgb[b] = FP(d_in, i++);
    }
    fgw = FP(d_in, i++); fgb = FP(d_in, i++);
  } else {
    for (int b = 0; b < 2; ++b) {
      bgb[b] = FP(d_in, i++); bgw[b] = FP(d_in, i++);
      g[b][1] = grag_srt(d_in, i);   // "inter" sorts before "intra"
      g[b][0] = grag_srt(d_in, i);
    }
    fgb = FP(d_in, i++); fgw = FP(d_in, i++); x = FP(d_in, i++);
  }

  // workspace layout
  float* wf = (float*)d_ws;
  Ws w; long off = 0;
  w.act = wf + off;    off += cBNPD;
  w.y   = wf + off;    off += cBNPD;
  w.u   = wf + off;    off += cBNPD;
  w.U   = wf + off;    off += cBNPD * 3 / 2;      // [bs, M, 3F]
  w.h   = wf + off;    off += cBNPD / 2;          // [bs, M, F] (also ffn2 hidden)
  w.pooled = wf + off; off += (long)cB * cN * cD;
  w.e1  = wf + off;    off += (long)cB * cN * cE;
  w.e2  = wf + off;    off += (long)cB * cN * cE;
  w.A1  = wf + off;    off += (long)cB * cN * cN;
  w.A2  = wf + off;    off += (long)cB * cN * cN;
  w.rsum = wf + off;   off += (long)cB * cN;
  w.csum = wf + off;   off += (long)cB * cN;

  float* outp = (float*)d_out;
  float* ATem = outp + cBNPD;                     // [B,P,P]
  float* ASpa = ATem + (long)cB * cP * cP;        // [B,N,N]

  const long PD = (long)cP * cD, ND = (long)cN * cD;
  const unsigned rows = (unsigned)(cB * cN * cP);
  const float* cur = x;
  for (int b = 0; b < 2; ++b) {
    // Intra: graph over patches. logical [B,M=P,S=N,D]; y layout = [B,N,P,D]
    run_grag(stream, cur, cP, cN, cD, PD, g[b][0], w.y, ATem, w);
    gate_kernel<<<rows, cD, 0, stream>>>(w.y, w.act, bgw[b], bgb[b],
                                         cN, cP, PD, cD, PD, cD);
    // Inter: graph over nodes. logical [B,M=N,S=P,D]; y layout = [B,P,N,D] -> transpose-gate
    run_grag(stream, w.act, cN, cP, (int)PD, cD, g[b][1], w.y, ASpa, w);
    gate_kernel<<<rows, cD, 0, stream>>>(w.y, w.act, bgw[b], bgb[b],
                                         cN, cP, cD, ND, PD, cD);
    cur = w.act;
  }
  // final gate -> d_out
  gate_kernel<<<rows, cD, 0, stream>>>(w.act, outp, fgw, fgb,
                                       cN, cP, PD, cD, PD, cD);
}